// MinBlcokScan_7507602833573
// MI455X (gfx1250) — compile-verified
//
#include <hip/hip_runtime.h>
#include <stdint.h>

#define L_LEN 65536
#define TILE  1024       // positions per block (8 waves x 2 subtiles x 64)
#define NTHR  256
#define NCH   8          // input channels per group, all staged in one pass
#define RSTR  1040       // floats per LDS row: 1032 used (4 halo + 1024 + 4 halo) + 8 pad

typedef __attribute__((ext_vector_type(2))) float v2f;
typedef __attribute__((ext_vector_type(8))) float v8f;

// CDNA5 async global->LDS copy (tracked by ASYNCcnt).
__device__ __forceinline__ void async_ld_b128(uint32_t lds_byte_off, const float* src) {
  uint64_t a = (uint64_t)src;
  asm volatile("global_load_async_to_lds_b128 %0, %1, off"
               :: "v"(lds_byte_off), "v"(a)
               : "memory");
}

__device__ __forceinline__ void wait_async0() {
#if __has_builtin(__builtin_amdgcn_s_wait_asynccnt)
  __builtin_amdgcn_s_wait_asynccnt(0);
#else
  asm volatile("s_wait_asynccnt 0" ::: "memory");
#endif
}

__global__ __launch_bounds__(NTHR) void grouped_conv1d_wmma_kernel(
    const float* __restrict__ x,
    const float* __restrict__ k0, const float* __restrict__ k1,
    const float* __restrict__ k2, const float* __restrict__ k3,
    float* __restrict__ y)
{
  __shared__ alignas(16) float lds[NCH * RSTR];     // 33,280 B input stage
  __shared__ alignas(16) float dsc[8][2][256];      // 16,384 B per-wave D scratch

  const int bid  = blockIdx.x;
  const int tile = bid & 63;          // 64 tiles of 1024 along L
  const int g    = (bid >> 6) & 3;    // 4 groups
  const int b    = bid >> 8;          // 32 batches
  const int tid  = threadIdx.x;
  const int pos0 = tile * TILE;

  const float* kg = (g == 0) ? k0 : (g == 1) ? k1 : (g == 2) ? k2 : k3;

  // ---- async stage: 8 channel rows, range clamped at tensor edges ----
  const int lo  = (pos0 == 0) ? 0 : pos0 - 4;
  const int hi  = (pos0 + TILE == L_LEN) ? L_LEN : pos0 + TILE + 4;
  const int nch = (hi - lo) >> 2;     // 16B chunks per channel row
  const int di0 = lo - pos0 + 4;      // lds float index of first staged element

  const float* xg = x + ((size_t)(b * 32 + g * 8)) * L_LEN;
  for (int r = 0; r < NCH; ++r) {
    const float* src = xg + (size_t)r * L_LEN + lo;
    for (int j = tid; j < nch; j += NTHR)
      async_ld_b128((uint32_t)((r * RSTR + di0 + 4 * j) * 4), src + 4 * j);
  }

  // zero conv-padding halo at tensor edges (disjoint from async-written range)
  if (tid < NCH) {
    if (pos0 == 0) {
#pragma unroll
      for (int q = 0; q < 4; ++q) lds[tid * RSTR + q] = 0.f;
    }
    if (pos0 + TILE == L_LEN) {
#pragma unroll
      for (int q = 0; q < 4; ++q) lds[tid * RSTR + TILE + 4 + q] = 0.f;
    }
  }

  wait_async0();
  __syncthreads();

  // ---- per-lane geometry ----
  const int lane = tid & 31;
  const int wv   = tid >> 5;          // wave id: subtile range [wv*128, wv*128+128)
  const int half = lane >> 4;         // lane half selects K pair (A/B) and M+8 (D)
  const int nn   = lane & 15;         // A row m / D column n
  const int o_a  = nn >> 2;           // A row -> (out-channel, phase)
  const int p_a  = nn & 3;

  // ---- build A (constant per group): 16 K-slices, v2f each ----
  // k = 4s + v + 2*half ; i = k>>3 ; t = (k&7)-2 ; A = K[o, i, t-p+2] or 0
  v2f Amat[16];
#pragma unroll
  for (int s = 0; s < 16; ++s) {
#pragma unroll
    for (int v = 0; v < 2; ++v) {
      int k = 4 * s + v + 2 * half;
      int i = k >> 3;
      int t = (k & 7) - 2;
      int w = t - p_a + 2;
      int wc = (w < 0) ? 0 : (w > 4 ? 4 : w);
      float tv = kg[(o_a * 8 + i) * 5 + wc];
      Amat[s][v] = (w == wc) ? tv : 0.f;
    }
  }

  // ---- WMMA main loop: 2 independent accumulator chains (2 subtiles) ----
  const int lb0 = wv * 128;
  const int lb1 = lb0 + 64;
  const int bdw = 4 * nn + 4;         // per-lane B column offset (+4 halo shift)

  v8f acc0 = {};
  v8f acc1 = {};
#pragma unroll
  for (int s = 0; s < 16; ++s) {
    v2f b0, b1;
#pragma unroll
    for (int v = 0; v < 2; ++v) {
      const int klo = 4 * s + v;
      const int khi = klo + 2;
      const int clo = (klo >> 3) * RSTR + ((klo & 7) - 2);
      const int chi = (khi >> 3) * RSTR + ((khi & 7) - 2);
      const int c   = half ? chi : clo;       // v_cndmask, compile-time constants
      b0[v] = lds[c + bdw + lb0];             // ds_load_b32
      b1[v] = lds[c + bdw + lb1];
    }
    acc0 = __builtin_amdgcn_wmma_f32_16x16x4_f32(false, Amat[s], false, b0,
                                                 (short)0, acc0, false, false);
    acc1 = __builtin_amdgcn_wmma_f32_16x16x4_f32(false, Amat[s], false, b1,
                                                 (short)0, acc1, false, false);
  }

  // ---- de-interleave phases via per-wave LDS scratch ----
#pragma unroll
  for (int v = 0; v < 8; ++v) {
    int mm = v + 8 * half;              // D row
    int o = mm >> 2, p = mm & 3;
    dsc[wv][0][o * 64 + 4 * nn + p] = acc0[v];
    dsc[wv][1][o * 64 + 4 * nn + p] = acc1[v];
  }
  // same-wave LDS ops are in-order: reads below see the writes above

  // ---- dense coalesced stores: 8 consecutive floats per lane ----
  {
    const int o_r = lane >> 3;          // lanes 0-7 -> ch0, ... 24-31 -> ch3
    const int c8  = lane & 7;
    float* dst0 = y + ((size_t)(b * 16 + g * 4 + o_r)) * L_LEN + pos0 + lb0 + c8 * 8;
    float* dst1 = y + ((size_t)(b * 16 + g * 4 + o_r)) * L_LEN + pos0 + lb1 + c8 * 8;
    const float4* rp0 = (const float4*)&dsc[wv][0][o_r * 64 + c8 * 8];
    const float4* rp1 = (const float4*)&dsc[wv][1][o_r * 64 + c8 * 8];
    float4 a0 = rp0[0], a1 = rp0[1];
    float4 b0v = rp1[0], b1v = rp1[1];
    ((float4*)dst0)[0] = a0;  ((float4*)dst0)[1] = a1;
    ((float4*)dst1)[0] = b0v; ((float4*)dst1)[1] = b1v;
  }
}

extern "C" void kernel_launch(void* const* d_in, const int* in_sizes, int n_in,
                              void* d_out, int out_size, void* d_ws, size_t ws_size,
                              hipStream_t stream) {
  const float* x  = (const float*)d_in[0];
  const float* k0 = (const float*)d_in[1];
  const float* k1 = (const float*)d_in[2];
  const float* k2 = (const float*)d_in[3];
  const float* k3 = (const float*)d_in[4];
  float* y = (float*)d_out;

  dim3 grid(32 * 4 * 64);   // batch x groups x L-tiles = 8192 blocks
  grouped_conv1d_wmma_kernel<<<grid, NTHR, 0, stream>>>(x, k0, k1, k2, k3, y);
}